// DistGen_34342558499035
// MI455X (gfx1250) — compile-verified
//
#include <hip/hip_runtime.h>
#include <hip/hip_bf16.h>
#include <stdint.h>

// ---------------------------------------------------------------------------
// Pointer-generator final-distribution kernel for MI455X (gfx1250).
//
// out[r, v] = p[r] * vocab_ds[r, v]                        (bulk, 824 MB stream)
// out[r, sources[l, r%B]] = p*vd + (1-p)*attns[r, l]       (sparse overwrite)
//
// Memory bound (0.125 FLOP/byte): optimize the stream, not the math.
// Bulk path: per-wave double-buffered async global->LDS copy
// (global_load_async_to_lds_b128 / s_wait_asynccnt), ds_load_b128 readback,
// non-temporal b128 stores.
// ---------------------------------------------------------------------------

typedef float v4f __attribute__((ext_vector_type(4)));   // native vector type

#define WAVE            32
#define WAVES_PER_BLOCK 8
#define BLOCK_THREADS   (WAVE * WAVES_PER_BLOCK)
#define TILE_BYTES      (WAVE * 16)   // 512 B per wave tile
#define TILE_FLOATS     (WAVE * 4)    // 128 floats per wave tile

__global__ void __launch_bounds__(BLOCK_THREADS)
DistGen_bulk_scale_kernel(const float* __restrict__ vocab_ds,
                          const float* __restrict__ p_gens,
                          float* __restrict__ out,
                          unsigned int V,
                          unsigned int numTiles)
{
    // 2 buffers * 8 waves * 512B = 8 KB LDS per block (WGP has 320 KB).
    __shared__ __align__(16) float lbuf[WAVES_PER_BLOCK][2][TILE_FLOATS];

    const unsigned lane = threadIdx.x & (WAVE - 1);
    const unsigned wib  = threadIdx.x / WAVE;                 // wave in block
    const unsigned numWaves = gridDim.x * WAVES_PER_BLOCK;

    unsigned tile = blockIdx.x * WAVES_PER_BLOCK + wib;       // uniform per wave
    if (tile >= numTiles) return;

    const v4f* curPtr = (const v4f*)&lbuf[wib][0][lane * 4];
    const v4f* nxtPtr = (const v4f*)&lbuf[wib][1][lane * 4];
    unsigned curAddr = (unsigned)(uintptr_t)curPtr;           // LDS byte offset
    unsigned nxtAddr = (unsigned)(uintptr_t)nxtPtr;

    // Prologue: async-copy first tile into buffer 0 (ASYNCcnt -> 1).
    {
        unsigned off = tile * TILE_BYTES + lane * 16u;
        asm volatile("global_load_async_to_lds_b128 %0, %1, %2"
                     :: "v"(curAddr), "v"(off), "s"(vocab_ds) : "memory");
    }

    for (;;) {
        const unsigned next   = tile + numWaves;
        const bool hasNext    = next < numTiles;              // wave-uniform

        if (hasNext) {
            // Prefetch next tile into the other buffer, then wait until only
            // that newest transfer is outstanding => current buffer is ready.
            unsigned off = next * TILE_BYTES + lane * 16u;
            asm volatile("global_load_async_to_lds_b128 %0, %1, %2"
                         :: "v"(nxtAddr), "v"(off), "s"(vocab_ds) : "memory");
            asm volatile("s_wait_asynccnt 0x1" ::: "memory");
        } else {
            asm volatile("s_wait_asynccnt 0x0" ::: "memory");
        }

        // Read current tile from LDS (ds_load_b128) and apply the row gate.
        v4f v = *curPtr;

        const unsigned i0  = tile * TILE_FLOATS + lane * 4u;  // flat element idx
        const unsigned r   = i0 / V;                          // row of first elem
        const unsigned bnd = (r + 1u) * V;                    // first idx of row r+1
        const float p0 = p_gens[r];
        // A float4 can straddle at most one row boundary (V >> 4); the
        // straddling element's row r+1 is guaranteed in-range.
        const float p1 = (i0 + 3u >= bnd) ? p_gens[r + 1u] : p0;

        v4f o;
        o.x = v.x * ((i0 + 0u < bnd) ? p0 : p1);
        o.y = v.y * ((i0 + 1u < bnd) ? p0 : p1);
        o.z = v.z * ((i0 + 2u < bnd) ? p0 : p1);
        o.w = v.w * ((i0 + 3u < bnd) ? p0 : p1);

        // Stream-once output: non-temporal 128-bit store.
        __builtin_nontemporal_store(o, (v4f*)(out + i0));

        if (!hasNext) break;
        tile = next;
        const v4f* tp = curPtr; curPtr = nxtPtr; nxtPtr = tp;
        unsigned   ta = curAddr; curAddr = nxtAddr; nxtAddr = ta;
    }
}

// Scalar tail for flat sizes not divisible by TILE_FLOATS (not hit for the
// reference shape: 2048*50257 is divisible by 128, but keep it general).
__global__ void DistGen_tail_scale_kernel(const float* __restrict__ vocab_ds,
                                          const float* __restrict__ p_gens,
                                          float* __restrict__ out,
                                          unsigned long long start,
                                          unsigned long long total,
                                          unsigned int V)
{
    unsigned long long i = start + blockIdx.x * (unsigned long long)blockDim.x
                         + threadIdx.x;
    if (i < total) {
        unsigned r = (unsigned)(i / V);
        out[i] = vocab_ds[i] * p_gens[r];
    }
}

// Sparse overwrite. One thread per output row; the serial l-loop makes
// duplicate-index resolution deterministic (last l wins), matching the
// reference's scatter-with-overwrite (duplicate order unspecified there).
// Must run after the bulk kernel on the same stream.
__global__ void DistGen_scatter_kernel(const float* __restrict__ vocab_ds,
                                       const float* __restrict__ attns,
                                       const float* __restrict__ p_gens,
                                       const int*   __restrict__ sources,
                                       float* __restrict__ out,
                                       unsigned rows, unsigned V,
                                       unsigned L, unsigned B)
{
    unsigned r = blockIdx.x * blockDim.x + threadIdx.x;
    if (r >= rows) return;

    const unsigned b = r % B;
    const float p = p_gens[r];
    const float g = 1.0f - p;
    const float* __restrict__ arow = attns + (size_t)r * L;
    const size_t obase = (size_t)r * V;

    for (unsigned l = 0; l < L; ++l) {
        unsigned v = (unsigned)sources[(size_t)l * B + b];
        if (v < V) {                       // mode="drop" semantics
            out[obase + v] = p * vocab_ds[obase + v] + g * arow[l];
        }
    }
}

extern "C" void kernel_launch(void* const* d_in, const int* in_sizes, int n_in,
                              void* d_out, int out_size, void* d_ws, size_t ws_size,
                              hipStream_t stream) {
    const float* vocab_ds = (const float*)d_in[0];   // (rows, V) fp32
    const float* attns    = (const float*)d_in[1];   // (rows, L) fp32
    const float* p_gens   = (const float*)d_in[2];   // (rows, 1) fp32
    const int*   sources  = (const int*)d_in[3];     // (L, B) int
    // d_in[4] = decoder_batch_len (scalar); not needed (rows/B derivable).
    float* out = (float*)d_out;

    const unsigned rows = (unsigned)in_sizes[2];                 // T*B
    const unsigned V    = (unsigned)(in_sizes[0] / in_sizes[2]);
    const unsigned L    = (unsigned)(in_sizes[1] / in_sizes[2]);
    const unsigned B    = (unsigned)(in_sizes[3] / (int)L);

    const unsigned long long total = (unsigned long long)rows * V;
    const unsigned numTiles = (unsigned)(total / TILE_FLOATS);   // full wave tiles
    const unsigned long long covered = (unsigned long long)numTiles * TILE_FLOATS;

    // 1) Bulk gated-vocab stream (99.9% of the 824 MB of traffic).
    if (numTiles > 0) {
        unsigned blocksNeeded = (numTiles + WAVES_PER_BLOCK - 1) / WAVES_PER_BLOCK;
        unsigned grid = blocksNeeded < 2048u ? blocksNeeded : 2048u;
        DistGen_bulk_scale_kernel<<<grid, BLOCK_THREADS, 0, stream>>>(
            vocab_ds, p_gens, out, V, numTiles);
    }

    // 2) Scalar tail (empty for the reference shape).
    if (covered < total) {
        unsigned rem = (unsigned)(total - covered);
        DistGen_tail_scale_kernel<<<(rem + 255) / 256, 256, 0, stream>>>(
            vocab_ds, p_gens, out, covered, total, V);
    }

    // 3) Sparse copy-distribution overwrite (after the bulk stream).
    DistGen_scatter_kernel<<<(rows + 255) / 256, 256, 0, stream>>>(
        vocab_ds, attns, p_gens, sources, out, rows, V, L, B);
}